// GIN_35880156791064
// MI455X (gfx1250) — compile-verified
//
#include <hip/hip_runtime.h>
#include <math.h>

typedef float v2f __attribute__((ext_vector_type(2)));
typedef float v8f __attribute__((ext_vector_type(8)));

constexpr int   kN     = 100000;  // nodes; 100000 = 1250 * 80
constexpr int   kE     = 800000;  // edges
constexpr float kEps   = 1e-5f;
constexpr int   MTILES = 5;       // 16-row M tiles per block -> 80 rows/block
constexpr int   ROWS   = MTILES * 16;

// ---------------------------------------------------------------- utilities

__global__ void k_zero(float* __restrict__ p, int n) {
  int i = blockIdx.x * blockDim.x + threadIdx.x;
  if (i < n) p[i] = 0.0f;
}

// --------------------------------------------- z init: z = pre(h) (elementwise)
// pre(h) = relu(h*scale + shift) when scale != nullptr (fused outer BN+ReLU),
// else identity. Vectorized float4.
__global__ void k_zinit4(float4* __restrict__ z, const float4* __restrict__ h,
                         const float* __restrict__ scale, const float* __restrict__ shift,
                         int n4, int Dm1) {
  int i = blockIdx.x * blockDim.x + threadIdx.x;
  if (i >= n4) return;
  float4 v = h[i];
  if (scale != nullptr) {
    int c = (i << 2) & Dm1;
    v.x = fmaxf(fmaf(v.x, scale[c + 0], shift[c + 0]), 0.0f);
    v.y = fmaxf(fmaf(v.y, scale[c + 1], shift[c + 1]), 0.0f);
    v.z = fmaxf(fmaf(v.z, scale[c + 2], shift[c + 2]), 0.0f);
    v.w = fmaxf(fmaf(v.w, scale[c + 3], shift[c + 3]), 0.0f);
  }
  z[i] = v;
}

// ------------------------------------------------------- edge scatter (agg)
// z[dst] += pre(h[src]); one thread per (edge, 4-feature chunk). Gathers and
// atomics stay in the 192MB L2 (h is 51MB).
__global__ void k_scatter(float* __restrict__ z, const float* __restrict__ h,
                          const int* __restrict__ src, const int* __restrict__ dst,
                          const float* __restrict__ scale, const float* __restrict__ shift,
                          int total, int chunks, int D) {
  int i = blockIdx.x * blockDim.x + threadIdx.x;
  if (i >= total) return;
  int e = i / chunks;
  int c = (i - e * chunks) << 2;
  int s = src[e];
  int d = dst[e];
  float4 v = *reinterpret_cast<const float4*>(h + (size_t)s * D + c);
  if (scale != nullptr) {
    v.x = fmaxf(fmaf(v.x, scale[c + 0], shift[c + 0]), 0.0f);
    v.y = fmaxf(fmaf(v.y, scale[c + 1], shift[c + 1]), 0.0f);
    v.z = fmaxf(fmaf(v.z, scale[c + 2], shift[c + 2]), 0.0f);
    v.w = fmaxf(fmaf(v.w, scale[c + 3], shift[c + 3]), 0.0f);
  }
  float* p = z + (size_t)d * D + c;
  atomicAdd(p + 0, v.x);
  atomicAdd(p + 1, v.y);
  atomicAdd(p + 2, v.z);
  atomicAdd(p + 3, v.w);
}

// ------------------------------------------------------------ fp32 WMMA GEMM
// Y[N x Dout] = preA(A)[N x Din] @ W[Din x Dout] + bias, with optional fused
// per-channel pre-op on A (BN+ReLU) and fused column sum/sumsq of Y.
// One block = 80 rows (5 M-tiles); wave w owns columns [16w,16w+16); each
// B fragment load feeds 5 v_wmma_f32_16x16x4_f32 issues.
// V_WMMA_F32_16X16X4_F32 layout (ISA 7.12.2):
//   A 16x4:  lane(0..15)=row M, VGPR j -> K = 2*(lane>>4)+j
//   B 4x16:  lane(0..15)=col N, VGPR j -> K = 2*(lane>>4)+j
//   C/D:     lanes 0-15 N=lane, VGPR v -> M=v; lanes 16-31 -> M=v+8
__global__ __launch_bounds__(256) void k_gemm_wmma(
    const float* __restrict__ A, const float* __restrict__ W,
    const float* __restrict__ bias, float* __restrict__ Y,
    const float* __restrict__ prescale, const float* __restrict__ preshift,
    float* __restrict__ sum, float* __restrict__ sumsq,
    int Din, int Dout) {
  __shared__ float sA[ROWS * 128];  // 40 KB

  const int tid  = threadIdx.x;
  const int wave = tid >> 5;
  const int lane = tid & 31;
  const int lm   = lane & 15;  // A row-in-tile / B,C column-in-tile
  const int kh   = lane >> 4;  // K half-select
  const int n0   = wave << 4;
  const int col  = n0 + lm;

  // ---- stage 80-row A panel into LDS (contiguous), optional BN+ReLU pre-op
  const int    tileElems = ROWS * Din;
  const size_t base      = (size_t)blockIdx.x * tileElems;
  const int    Dm1       = Din - 1;
  if (prescale != nullptr) {
    for (int i = tid; i < tileElems; i += blockDim.x) {
      int   c = i & Dm1;
      float v = fmaf(A[base + i], prescale[c], preshift[c]);
      sA[i]   = fmaxf(v, 0.0f);
    }
  } else {
    for (int i = tid; i < tileElems; i += blockDim.x) sA[i] = A[base + i];
  }
  __syncthreads();

  // ---- K loop: one B fragment -> 5 WMMAs
  v8f acc[MTILES] = {};
  for (int k0 = 0; k0 < Din; k0 += 4) {
    const int ka = k0 + (kh << 1);
    v2f b;
    b.x = W[ka * Dout + col];
    b.y = W[(ka + 1) * Dout + col];
#pragma unroll
    for (int t = 0; t < MTILES; ++t) {
      const int r = (t << 4) + lm;
      v2f a;
      a.x = sA[r * Din + ka];
      a.y = sA[r * Din + ka + 1];
      acc[t] = __builtin_amdgcn_wmma_f32_16x16x4_f32(
          false, a, false, b, (short)0, acc[t], false, false);
    }
  }

  // ---- epilogue: bias, store, fused column statistics
  const float bv = bias[col];
  float s = 0.0f, s2 = 0.0f;
#pragma unroll
  for (int t = 0; t < MTILES; ++t) {
    const int r0 = blockIdx.x * ROWS + (t << 4) + (kh << 3);
#pragma unroll
    for (int v = 0; v < 8; ++v) {
      float val = acc[t][v] + bv;
      Y[(size_t)(r0 + v) * Dout + col] = val;
      s  += val;
      s2 += val * val;
    }
  }
  if (sum != nullptr) {
    s  += __shfl_xor(s, 16);   // combine kh=0 / kh=1 partners (wave32)
    s2 += __shfl_xor(s2, 16);
    if (kh == 0) {
      atomicAdd(&sum[col], s);
      atomicAdd(&sumsq[col], s2);
    }
  }
}

// Fold mean/var/gamma/beta into per-channel affine: y*scale + shift
__global__ void k_bnparams(const float* __restrict__ sum, const float* __restrict__ sumsq,
                           const float* __restrict__ g, const float* __restrict__ be,
                           float* __restrict__ scale, float* __restrict__ shift,
                           int D, float invN) {
  int c = blockIdx.x * blockDim.x + threadIdx.x;
  if (c >= D) return;
  float m   = sum[c] * invN;
  float var = fmaf(-m, m, sumsq[c] * invN);  // biased variance (torch-style)
  float a   = g[c] * rsqrtf(var + kEps);
  scale[c]  = a;
  shift[c]  = fmaf(-m, a, be[c]);
}

// ---------------------------------------------------------------- host side

extern "C" void kernel_launch(void* const* d_in, const int* in_sizes, int n_in,
                              void* d_out, int out_size, void* d_ws, size_t ws_size,
                              hipStream_t stream) {
  (void)in_sizes; (void)n_in; (void)out_size; (void)ws_size;

  const float* x    = (const float*)d_in[0];
  const int*   ei   = (const int*)d_in[1];
  const int*   srcI = ei;        // edge_index[0]
  const int*   dstI = ei + kE;   // edge_index[1]

  const float* W1[3] = {(const float*)d_in[2],  (const float*)d_in[8],  (const float*)d_in[14]};
  const float* b1[3] = {(const float*)d_in[3],  (const float*)d_in[9],  (const float*)d_in[15]};
  const float* g1[3] = {(const float*)d_in[4],  (const float*)d_in[10], (const float*)d_in[16]};
  const float* e1[3] = {(const float*)d_in[5],  (const float*)d_in[11], (const float*)d_in[17]};
  const float* W2[3] = {(const float*)d_in[6],  (const float*)d_in[12], (const float*)d_in[18]};
  const float* b2[3] = {(const float*)d_in[7],  (const float*)d_in[13], (const float*)d_in[19]};
  const float* go[2] = {(const float*)d_in[20], (const float*)d_in[22]};
  const float* bo[2] = {(const float*)d_in[21], (const float*)d_in[23]};

  float* out = (float*)d_out;

  float* ws    = (float*)d_ws;
  size_t NB    = (size_t)kN * 128;
  float* bufZ  = ws;            // aggregated input z
  float* bufY1 = ws + NB;       // inner GEMM output
  float* bufY2 = ws + 2 * NB;   // outer GEMM output (pre-BN h)
  float* sumA  = ws + 3 * NB;   // inner BN stats/params
  float* sqA   = sumA + 128;
  float* sclA  = sqA + 128;
  float* shfA  = sclA + 128;
  float* sumB  = shfA + 128;    // outer BN stats/params
  float* sqB   = sumB + 128;
  float* sclB  = sqB + 128;
  float* shfB  = sclB + 128;

  const int TB = 256;
  const int GB = kN / ROWS;  // 1250 exact GEMM blocks

  // z = pre(h); z[dst] += pre(h[src])   (pre = outer BN+ReLU of prev layer)
  auto aggregate = [&](const float* h, const float* scl, const float* shf, int D) {
    int n4 = (kN * D) >> 2;
    k_zinit4<<<(n4 + TB - 1) / TB, TB, 0, stream>>>((float4*)bufZ, (const float4*)h,
                                                    scl, shf, n4, D - 1);
    int chunks = D >> 2;
    int total  = kE * chunks;
    k_scatter<<<(total + TB - 1) / TB, TB, 0, stream>>>(bufZ, h, srcI, dstI,
                                                        scl, shf, total, chunks, D);
  };

  for (int l = 0; l < 3; ++l) {
    const int Din = (l == 0) ? 64 : 128;
    if (l == 0)
      aggregate(x, nullptr, nullptr, 64);         // raw input, no pre-op
    else
      aggregate(bufY2, sclB, shfB, 128);          // fused outer BN+ReLU of prev layer

    // inner GEMM: y1 = z @ W1 + b1, fused column stats for inner BN
    k_zero<<<1, 256, 0, stream>>>(sumA, 256);
    k_gemm_wmma<<<GB, 256, 0, stream>>>(bufZ, W1[l], b1[l], bufY1,
                                        nullptr, nullptr, sumA, sqA, Din, 128);
    k_bnparams<<<1, 128, 0, stream>>>(sumA, sqA, g1[l], e1[l], sclA, shfA,
                                      128, 1.0f / kN);

    if (l < 2) {
      // outer GEMM: y2 = relu(bn(y1)) @ W2 + b2, fused stats for outer BN
      k_zero<<<1, 256, 0, stream>>>(sumB, 256);
      k_gemm_wmma<<<GB, 256, 0, stream>>>(bufY1, W2[l], b2[l], bufY2,
                                          sclA, shfA, sumB, sqB, 128, 128);
      k_bnparams<<<1, 128, 0, stream>>>(sumB, sqB, go[l], bo[l], sclB, shfB,
                                        128, 1.0f / kN);
    } else {
      // final GEMM: out = relu(bn(y1)) @ W2 + b2  (no outer norm, Dout=64)
      k_gemm_wmma<<<GB, 128, 0, stream>>>(bufY1, W2[l], b2[l], out,
                                          sclA, shfA, nullptr, nullptr, 128, 64);
    }
  }
}